// TensorizedEmbedding_27169963114596
// MI455X (gfx1250) — compile-verified
//
#include <hip/hip_runtime.h>

// ---------------------------------------------------------------------------
// TT embedding:  x in [0,32000) = digits (d0,d1,d2,d3) over radices (8,10,20,20)
//   core0: (1, 8,4,16)   core1: (16,10,4,16)   core2: (16,20,4,16)  core3: (16,20,8,1)
//   out row j = e*128 + m1*32 + m2*8 + m3   (512 floats, fp32)
//
// Plan: build the full 32000x512 table (62.5 MB -> fits the 192 MB L2) with
// V_WMMA_F32_16X16X4_F32, one wave per (d0,d1,d2) prefix (1600 combos, d3
// reused 20x), then a pure gather with NON-TEMPORAL output stores so the
// 256 MB output stream doesn't evict the table from L2.
// ---------------------------------------------------------------------------

typedef float v2f __attribute__((ext_vector_type(2)));
typedef float v4f __attribute__((ext_vector_type(4)));
typedef float v8f __attribute__((ext_vector_type(8)));

static __device__ __forceinline__ v8f wmma_f32(v2f a, v2f b, v8f c) {
  // (neg_a, A, neg_b, B, c_mod, C, reuse_a, reuse_b)
  return __builtin_amdgcn_wmma_f32_16x16x4_f32(false, a, false, b, (short)0, c,
                                               false, false);
}

// One wave computes one (d0,d1,d2) combo and writes 20 table rows (d3 = 0..19).
__global__ __launch_bounds__(256) void tt_build_table(
    const float* __restrict__ core0, const float* __restrict__ core1,
    const float* __restrict__ core2, const float* __restrict__ core3,
    float* __restrict__ tbl) {
  __shared__ float lds[8][1280];  // per wave: M1 16x16 (256) + M2 64x16 (1024)

  const int lane  = threadIdx.x & 31;
  const int wave  = threadIdx.x >> 5;
  const int combo = blockIdx.x * 8 + wave;  // 0..1599 exactly (grid = 200)
  const int d2 = combo % 20;
  const int t  = combo / 20;
  const int d1 = t % 10;
  const int d0 = t / 10;

  const int m     = lane & 15;         // A: M row; B: N column
  const int khalf = (lane >> 4) * 2;   // K sub-offset within a 4-wide WMMA step
  const int rhalf = (lane >> 4) * 8;   // C/D: +8 row offset for upper lane half

  float* M1 = &lds[wave][0];    // R1 reshaped to 16x16  (row = e*4+m1)
  float* M2 = &lds[wave][256];  // R2 reshaped to 64x16  (row = row1*4+m2)

  // ---- Stage 1: R1(4x64) = v0(4x16) @ B1(16x64),  B1[r][m1*16+s]=core1[r,d1,m1,s]
  // A fragments: unconditional loads from a clamped (always-valid) row, then a
  // register select for the zero padding -> no exec-mask branches around WMMA.
  v2f a1[4];
#pragma unroll
  for (int kk = 0; kk < 4; ++kk) {
    const int k = kk * 4 + khalf;
    const float x0 = core0[(d0 * 4 + (m & 3)) * 16 + k];
    const float x1 = core0[(d0 * 4 + (m & 3)) * 16 + k + 1];
    a1[kk].x = (m < 4) ? x0 : 0.f;
    a1[kk].y = (m < 4) ? x1 : 0.f;
  }
#pragma unroll
  for (int n = 0; n < 4; ++n) {  // N-tile n == m1
    v8f acc = {0.f, 0.f, 0.f, 0.f, 0.f, 0.f, 0.f, 0.f};
#pragma unroll
    for (int kk = 0; kk < 4; ++kk) {
      const int k = kk * 4 + khalf;
      v2f b;
      b.x = core1[((k * 10 + d1) * 4 + n) * 16 + m];
      b.y = core1[(((k + 1) * 10 + d1) * 4 + n) * 16 + m];
      acc = wmma_f32(a1[kk], b, acc);
    }
    // R1'[e*4+n][s] = acc[e][s]; rows e=0..3 live in VGPRs 0..3, lanes 0-15.
    if (lane < 16) {
#pragma unroll
      for (int g = 0; g < 4; ++g) M1[(g * 4 + n) * 16 + lane] = acc[g];
    }
  }

  // ---- Stage 2: R2(16x64) = R1'(16x16) @ B2(16x64), B2[r][m2*16+s]=core2[r,d2,m2,s]
  v2f a2[4];
#pragma unroll
  for (int kk = 0; kk < 4; ++kk) {
    const int k = kk * 4 + khalf;
    a2[kk].x = M1[m * 16 + k];
    a2[kk].y = M1[m * 16 + k + 1];
  }
#pragma unroll
  for (int n2 = 0; n2 < 4; ++n2) {  // N-tile n2 == m2
    v8f acc = {0.f, 0.f, 0.f, 0.f, 0.f, 0.f, 0.f, 0.f};
#pragma unroll
    for (int kk = 0; kk < 4; ++kk) {
      const int k = kk * 4 + khalf;
      v2f b;
      b.x = core2[((k * 20 + d2) * 4 + n2) * 16 + m];
      b.y = core2[(((k + 1) * 20 + d2) * 4 + n2) * 16 + m];
      acc = wmma_f32(a2[kk], b, acc);
    }
    // R2'[row1*4+n2][s] = acc[row1][s];  row1 = g (+8 for upper lane half)
#pragma unroll
    for (int g = 0; g < 8; ++g) M2[((g + rhalf) * 4 + n2) * 16 + m] = acc[g];
  }

  // ---- Stage 3: for each d3: R3(64x8) = R2'(64x16) @ B3(16x8),
  //               B3[r][m3] = core3[r,d3,m3,0]
  v2f a3[4][4];  // preload A fragments once; reused across all 20 d3
#pragma unroll
  for (int mt = 0; mt < 4; ++mt)
#pragma unroll
    for (int kk = 0; kk < 4; ++kk) {
      const int k = kk * 4 + khalf;
      a3[mt][kk].x = M2[(mt * 16 + m) * 16 + k];
      a3[mt][kk].y = M2[(mt * 16 + m) * 16 + k + 1];
    }

  const long long xbase = (long long)combo * 20;
  for (int d3 = 0; d3 < 20; ++d3) {
    v2f b3[4];
#pragma unroll
    for (int kk = 0; kk < 4; ++kk) {
      const int k = kk * 4 + khalf;
      const float y0 = core3[(k * 20 + d3) * 8 + (m & 7)];        // always valid
      const float y1 = core3[((k + 1) * 20 + d3) * 8 + (m & 7)];  // always valid
      b3[kk].x = (m < 8) ? y0 : 0.f;
      b3[kk].y = (m < 8) ? y1 : 0.f;
    }
    float* __restrict__ row = tbl + (xbase + d3) * 512;
#pragma unroll
    for (int mt = 0; mt < 4; ++mt) {
      v8f acc = {0.f, 0.f, 0.f, 0.f, 0.f, 0.f, 0.f, 0.f};
#pragma unroll
      for (int kk = 0; kk < 4; ++kk) acc = wmma_f32(a3[mt][kk], b3[kk], acc);
      if (m < 8) {  // N columns 8..15 are padding
#pragma unroll
        for (int g = 0; g < 8; ++g) {
          const int row2 = mt * 16 + g + rhalf;
          row[row2 * 8 + m] = acc[g];
        }
      }
    }
  }
}

// Pure gather from the L2-resident table: out[i][:] = tbl[x[i]][:].
// Output is streamed with NT stores so it does not evict the table from L2.
__global__ __launch_bounds__(256) void tt_gather(const int* __restrict__ x,
                                                 const v4f* __restrict__ tbl4,
                                                 v4f* __restrict__ out4,
                                                 long long total4) {
  const long long gid = (long long)blockIdx.x * 256 + threadIdx.x;
  if (gid >= total4) return;
  const long long rowi = gid >> 7;  // 128 float4 per 512-float row
  const int col = (int)(gid & 127);
  const int xv = x[rowi];
  const v4f v = tbl4[(long long)xv * 128 + col];  // RT load: keep table hot
  __builtin_nontemporal_store(v, &out4[gid]);     // NT store: bypass L2 reuse
}

// Fallback (only if ws_size cannot hold the 62.5 MB table): direct scalar compute.
__global__ __launch_bounds__(64) void tt_direct(
    const int* __restrict__ x, const float* __restrict__ core0,
    const float* __restrict__ core1, const float* __restrict__ core2,
    const float* __restrict__ core3, float* __restrict__ out) {
  const int rowi = blockIdx.x;
  const int row2 = threadIdx.x;  // 0..63 = e*16 + m1*4 + m2
  const int xv = x[rowi];
  const int d0 = (xv / 4000) % 8, d1 = (xv / 400) % 10;
  const int d2 = (xv / 20) % 20, d3 = xv % 20;
  const int e = row2 >> 4, m1 = (row2 >> 2) & 3, m2 = row2 & 3;
  float r1[16], r2[16];
  for (int s = 0; s < 16; ++s) {
    float acc = 0.f;
    for (int r = 0; r < 16; ++r)
      acc += core0[(d0 * 4 + e) * 16 + r] * core1[((r * 10 + d1) * 4 + m1) * 16 + s];
    r1[s] = acc;
  }
  for (int s = 0; s < 16; ++s) {
    float acc = 0.f;
    for (int r = 0; r < 16; ++r)
      acc += r1[r] * core2[((r * 20 + d2) * 4 + m2) * 16 + s];
    r2[s] = acc;
  }
  for (int m3 = 0; m3 < 8; ++m3) {
    float acc = 0.f;
    for (int r = 0; r < 16; ++r) acc += r2[r] * core3[(r * 20 + d3) * 8 + m3];
    out[(long long)rowi * 512 + row2 * 8 + m3] = acc;
  }
}

extern "C" void kernel_launch(void* const* d_in, const int* in_sizes, int n_in,
                              void* d_out, int out_size, void* d_ws,
                              size_t ws_size, hipStream_t stream) {
  const int* x = (const int*)d_in[0];
  const float* c0 = (const float*)d_in[1];
  const float* c1 = (const float*)d_in[2];
  const float* c2 = (const float*)d_in[3];
  const float* c3 = (const float*)d_in[4];
  float* out = (float*)d_out;
  const int n = in_sizes[0];  // 131072 index lookups

  const size_t tbl_bytes = 32000ull * 512ull * sizeof(float);  // 62.5 MB
  if (ws_size >= tbl_bytes) {
    float* tbl = (float*)d_ws;
    tt_build_table<<<200, 256, 0, stream>>>(c0, c1, c2, c3, tbl);  // 1600 combos
    const long long total4 = (long long)n * 128;
    const int blocks = (int)((total4 + 255) / 256);
    tt_gather<<<blocks, 256, 0, stream>>>(x, (const v4f*)tbl, (v4f*)out, total4);
  } else {
    tt_direct<<<n, 64, 0, stream>>>(x, c0, c1, c2, c3, out);
  }
}